// SSMInterBlock_54236847014470
// MI455X (gfx1250) — compile-verified
//
#include <hip/hip_runtime.h>
#include <hip/hip_bf16.h>

typedef __attribute__((ext_vector_type(2))) float v2f;
typedef __attribute__((ext_vector_type(8))) float v8f;

#define XZ_STRIDE 385           // 384 cols + 1 pad (bank-conflict mitigation)
#define NPIX_PER_WAVE 4
#define WAVES_PER_BLOCK 2

struct WaveSmem {
    float xz[16 * XZ_STRIDE];   // post-SiLU [x1 | z]; x1 half later overwritten by y, then yz
    float xdbl[16 * 48];        // dts[0:6), B[6:22), C[22:38), pad to 48
    float mu[16];
    float rs[16];
};

__global__ __launch_bounds__(64) void
ssm_fused_kernel(const float* __restrict__ x,        // [npix, 96, 4] (d-major, l inner)
                 const float* __restrict__ Win,      // [384, 96]
                 const float* __restrict__ Wxp,      // [38, 192]
                 const float* __restrict__ Wdt,      // [192, 6]
                 const float* __restrict__ dt_bias,  // [192]
                 const float* __restrict__ A_logs,   // [192, 16]
                 const float* __restrict__ Dvec,     // [192]
                 const float* __restrict__ gamma,    // [192]
                 const float* __restrict__ beta,     // [192]
                 const float* __restrict__ Wout,     // [96, 192]
                 float* __restrict__ out,            // [npix, 96, 4]
                 int npix)
{
    __shared__ WaveSmem smem[WAVES_PER_BLOCK];

    const int wave = threadIdx.x >> 5;
    const int lane = threadIdx.x & 31;
    const int pix0 = (blockIdx.x * WAVES_PER_BLOCK + wave) * NPIX_PER_WAVE;
    if (pix0 + NPIX_PER_WAVE > npix) return;   // grid is exact: npix % 8 == 0

    WaveSmem& S = smem[wave];

    // WMMA fp32 16x16x4 fragment mapping (ISA 7.12.2):
    //  A: lanes 0-15 hold row M=lane, K = {4k,4k+1}; lanes 16-31 row M=lane-16, K = {4k+2,4k+3}
    //  D: lanes 0-15 hold col N=lane, VGPR r -> M=r; lanes 16-31 col N=lane-16, M=r+8
    const int rowM  = lane & 15;
    const int khalf = lane >> 4;          // 0 or 1
    // logical row r -> (pixel pix0 + r/4, l = r&3); x element (row, k) at x[(pix)*384 + k*4 + l]
    const float* xrow = x + (size_t)(pix0 + (rowM >> 2)) * 384 + (rowM & 3);

    // ---------------- Stage 1: GEMM1 + SiLU: xz = silu(xf @ Win^T), N=384 ----------------
    for (int nt = 0; nt < 24; ++nt) {
        const int c0 = nt * 16;
        v8f acc = {};
        const float* wb = Win + (size_t)(c0 + rowM) * 96 + 2 * khalf;
        for (int kk = 0; kk < 24; ++kk) {
            const int kb = kk * 4 + 2 * khalf;
            v2f a; a.x = xrow[(size_t)kb * 4]; a.y = xrow[(size_t)(kb + 1) * 4];
            v2f b; b.x = wb[kk * 4];           b.y = wb[kk * 4 + 1];
            acc = __builtin_amdgcn_wmma_f32_16x16x4_f32(false, a, false, b,
                                                        (short)0, acc, false, false);
        }
        const int ncol = c0 + rowM;
        for (int r = 0; r < 8; ++r) {
            float v = acc[r];
            v = v / (1.0f + __expf(-v));                       // SiLU
            S.xz[(r + 8 * khalf) * XZ_STRIDE + ncol] = v;
        }
    }
    __syncthreads();

    // ---------------- Stage 2: x_dbl = u @ Wxp^T, N=38 (padded to 48) ----------------
    for (int nt = 0; nt < 3; ++nt) {
        const int c0 = nt * 16;
        const int c  = c0 + rowM;
        const bool valid = (c < 38);
        const float* wb = valid ? (Wxp + (size_t)c * 192 + 2 * khalf) : Wxp;
        v8f acc = {};
        const float* arow = &S.xz[rowM * XZ_STRIDE];
        for (int kk = 0; kk < 48; ++kk) {
            const int kb = kk * 4 + 2 * khalf;
            v2f a; a.x = arow[kb]; a.y = arow[kb + 1];
            v2f b;
            b.x = valid ? wb[kk * 4]     : 0.0f;
            b.y = valid ? wb[kk * 4 + 1] : 0.0f;
            acc = __builtin_amdgcn_wmma_f32_16x16x4_f32(false, a, false, b,
                                                        (short)0, acc, false, false);
        }
        for (int r = 0; r < 8; ++r)
            S.xdbl[(r + 8 * khalf) * 48 + c0 + rowM] = acc[r];
    }
    __syncthreads();

    // ---------------- Stage 3: delta + softplus + selective scan (L=4, N=16) ----------------
    // lane <-> one channel (ch = it*32 + lane, 6 iterations); pixels iterated sequentially.
    // Per-channel params (Av, Wdt, bias, D) computed ONCE per channel (no 4x pixel redundancy).
    {
        float lnsum[16], lnsq[16];
        #pragma unroll
        for (int r = 0; r < 16; ++r) { lnsum[r] = 0.f; lnsq[r] = 0.f; }

        for (int it = 0; it < 6; ++it) {
            const int ch = it * 32 + lane;
            float Av[16];
            #pragma unroll
            for (int n = 0; n < 16; ++n) Av[n] = -__expf(A_logs[ch * 16 + n]);
            const float dtb = dt_bias[ch];
            const float dch = Dvec[ch];
            float wdt[6];
            #pragma unroll
            for (int r = 0; r < 6; ++r) wdt[r] = Wdt[ch * 6 + r];

            #pragma unroll
            for (int p = 0; p < 4; ++p) {
                float h[16];
                #pragma unroll
                for (int n = 0; n < 16; ++n) h[n] = 0.f;
                #pragma unroll
                for (int l = 0; l < 4; ++l) {
                    const int row = p * 4 + l;
                    const float* xd = &S.xdbl[row * 48];   // wave-uniform LDS broadcast
                    float dacc = dtb;
                    #pragma unroll
                    for (int r = 0; r < 6; ++r) dacc += xd[r] * wdt[r];
                    const float delta = (dacc > 20.f) ? dacc
                                       : __logf(1.0f + __expf(dacc));   // softplus
                    const float u  = S.xz[row * XZ_STRIDE + ch];
                    const float du = delta * u;
                    float y = 0.f;
                    #pragma unroll
                    for (int n = 0; n < 16; ++n) {
                        const float dA = __expf(delta * Av[n]);
                        h[n] = dA * h[n] + du * xd[6 + n];
                        y += h[n] * xd[22 + n];
                    }
                    y += dch * u;
                    S.xz[row * XZ_STRIDE + ch] = y;   // overwrite x1 with y (same lane, same slot)
                    lnsum[row] += y; lnsq[row] += y * y;
                }
            }
        }
        // full-wave butterfly: every lane ends with per-row totals over all 192 channels
        #pragma unroll
        for (int m = 1; m < 32; m <<= 1) {
            #pragma unroll
            for (int r = 0; r < 16; ++r) {
                lnsum[r] += __shfl_xor(lnsum[r], m, 32);
                lnsq[r]  += __shfl_xor(lnsq[r],  m, 32);
            }
        }
        if (lane == 0) {
            #pragma unroll
            for (int r = 0; r < 16; ++r) {
                const float mu  = lnsum[r] * (1.0f / 192.0f);
                const float var = lnsq[r] * (1.0f / 192.0f) - mu * mu;
                S.mu[r] = mu;
                S.rs[r] = rsqrtf(var + 1e-5f);
            }
        }
    }
    __syncthreads();

    // ---------------- Stage 4: y = LN(y) * gamma + beta, then * z (in place) ----------------
    for (int i = lane; i < 16 * 192; i += 32) {
        const int row = i / 192;
        const int ch  = i - row * 192;
        const float y = S.xz[row * XZ_STRIDE + ch];
        float v = (y - S.mu[row]) * S.rs[row] * gamma[ch] + beta[ch];
        v *= S.xz[row * XZ_STRIDE + 192 + ch];        // * z
        S.xz[row * XZ_STRIDE + ch] = v;
    }
    __syncthreads();

    // ---------------- Stage 5: GEMM2: out = yz @ Wout^T, N=96, K=192 ----------------
    for (int nt = 0; nt < 6; ++nt) {
        const int c0 = nt * 16;
        v8f acc = {};
        const float* wb = Wout + (size_t)(c0 + rowM) * 192 + 2 * khalf;
        const float* arow = &S.xz[rowM * XZ_STRIDE];
        for (int kk = 0; kk < 48; ++kk) {
            const int kb = kk * 4 + 2 * khalf;
            v2f a; a.x = arow[kb];   a.y = arow[kb + 1];
            v2f b; b.x = wb[kk * 4]; b.y = wb[kk * 4 + 1];
            acc = __builtin_amdgcn_wmma_f32_16x16x4_f32(false, a, false, b,
                                                        (short)0, acc, false, false);
        }
        const int ncol = c0 + rowM;
        for (int r = 0; r < 8; ++r) {
            const int row = r + 8 * khalf;
            out[(size_t)(pix0 + (row >> 2)) * 384 + ncol * 4 + (row & 3)] = acc[r];
        }
    }
}

extern "C" void kernel_launch(void* const* d_in, const int* in_sizes, int n_in,
                              void* d_out, int out_size, void* d_ws, size_t ws_size,
                              hipStream_t stream) {
    const float* x       = (const float*)d_in[0];
    const float* Win     = (const float*)d_in[1];
    const float* Wxp     = (const float*)d_in[2];
    const float* Wdt     = (const float*)d_in[3];
    const float* dt_bias = (const float*)d_in[4];
    const float* A_logs  = (const float*)d_in[5];
    const float* Dvec    = (const float*)d_in[6];
    const float* gamma   = (const float*)d_in[7];
    const float* beta    = (const float*)d_in[8];
    const float* Wout    = (const float*)d_in[9];
    float* out = (float*)d_out;

    const int npix = in_sizes[0] / (96 * 4);          // 12544
    const int pix_per_block = NPIX_PER_WAVE * WAVES_PER_BLOCK;  // 8
    const int grid = npix / pix_per_block;            // 1568 (npix divisible by 8)

    ssm_fused_kernel<<<grid, 64, 0, stream>>>(x, Win, Wxp, Wdt, dt_bias, A_logs,
                                              Dvec, gamma, beta, Wout, out, npix);
}